// GAT_10170482556988
// MI455X (gfx1250) — compile-verified
//
#include <hip/hip_runtime.h>
#include <math.h>

#define HC 128
#define HEADS 2
#define BGRAPHS 64
#define EPSBN 1e-5f
#define NEG_SLOPE 0.2f
#define MBLK 4   // M-tiles per wave in the WMMA GEMM

typedef float v2f __attribute__((ext_vector_type(2)));
typedef float v8f __attribute__((ext_vector_type(8)));

// ---------------------------------------------------------------- utilities
__global__ void fill_f32(float* __restrict__ p, float v, long long n) {
  long long i = (long long)blockIdx.x * blockDim.x + threadIdx.x;
  if (i < n) p[i] = v;
}

__device__ __forceinline__ float leaky(float x) {
  return x > 0.f ? x : NEG_SLOPE * x;
}

// float atomic max via int/uint ordering trick (init must be -inf)
__device__ __forceinline__ void atomicMaxF(float* addr, float val) {
  if (val >= 0.f) atomicMax((int*)addr, __float_as_int(val));
  else            atomicMin((unsigned int*)addr, (unsigned int)__float_as_int(val));
}

// ------------------------------------------------ fp32 WMMA GEMM  D = A @ B
// A[M,K] row-major, B[K,NC] row-major, D[M,NC]. NC%16==0, K%4==0.
// One wave computes a 64x16 slab (MBLK=4 tiles of 16x16) with
// V_WMMA_F32_16X16X4_F32: the B fragment is loaded once per K-step and
// reused by 4 independent WMMAs (amortizes loads, hides XDL latency).
__global__ void gemm_wmma_f32(const float* __restrict__ A,
                              const float* __restrict__ B,
                              float* __restrict__ D, int M, int K, int NC) {
  int wave = (int)(((long long)blockIdx.x * blockDim.x + threadIdx.x) >> 5);
  int lane = threadIdx.x & 31;
  int ntiles = NC >> 4;
  int mblk = wave / ntiles;
  int nt   = wave - mblk * ntiles;
  int mt0  = mblk * MBLK;
  if (mt0 * 16 >= M) return;                // uniform per wave -> EXEC all 1s

  int r  = lane & 15;                       // m index for A, n index for B/D
  int kh = lane >> 4;                       // k half (K=0,1 vs K=2,3)

  const float* bcol = B + nt * 16 + r;
  const float* arow[MBLK];
  bool valid[MBLK];
#pragma unroll
  for (int i = 0; i < MBLK; ++i) {
    valid[i] = (mt0 + i) * 16 < M;
    int row = (mt0 + i) * 16 + r;
    if (row > M - 1) row = M - 1;           // clamp: tail reads valid memory
    arow[i] = A + (long long)row * K;
  }

  v8f acc[MBLK] = {};
  for (int k0 = 0; k0 < K; k0 += 4) {
    int k = k0 + 2 * kh;
    v2f b;
    b.x = bcol[(long long)k * NC];
    b.y = bcol[(long long)(k + 1) * NC];
#pragma unroll
    for (int i = 0; i < MBLK; ++i) {
      v2f a;
      a.x = arow[i][k];
      a.y = arow[i][k + 1];
      acc[i] = __builtin_amdgcn_wmma_f32_16x16x4_f32(false, a, false, b,
                                                     (short)0, acc[i], false, false);
    }
  }
  // C/D layout: VGPR j -> M = j (lanes 0-15) or j+8 (lanes 16-31), N = lane&15
#pragma unroll
  for (int i = 0; i < MBLK; ++i) {
    if (!valid[i]) continue;
    float* drow = D + (long long)((mt0 + i) * 16 + 8 * kh) * NC + nt * 16 + r;
#pragma unroll
    for (int j = 0; j < 8; ++j)
      drow[(long long)j * NC] = acc[i][j];
  }
}

// --------------------------------------- per-node attention scalars (one wave)
__global__ void node_alpha(const float* __restrict__ h,
                           const float* __restrict__ a_src,
                           const float* __restrict__ a_dst,
                           float* __restrict__ as_, float* __restrict__ ad_,
                           int n) {
  int wave = (int)(((long long)blockIdx.x * blockDim.x + threadIdx.x) >> 5);
  int lane = threadIdx.x & 31;
  if (wave >= n) return;
  float4 hv = ((const float4*)(h + (long long)wave * HC))[lane];
  float4 s4 = ((const float4*)a_src)[lane];   // [H,C] flat matches channel order
  float4 d4 = ((const float4*)a_dst)[lane];
  float s = hv.x * s4.x + hv.y * s4.y + hv.z * s4.z + hv.w * s4.w;
  float d = hv.x * d4.x + hv.y * d4.y + hv.z * d4.z + hv.w * d4.w;
#pragma unroll
  for (int off = 8; off >= 1; off >>= 1) {    // reduce within 16-lane head group
    s += __shfl_xor(s, off, 32);
    d += __shfl_xor(d, off, 32);
  }
  if ((lane & 15) == 0) {
    int head = lane >> 4;
    as_[wave * HEADS + head] = s;
    ad_[wave * HEADS + head] = d;
  }
}

// ------------------------------------------------ softmax pass 1: segment max
__global__ void edge_max(const int* __restrict__ ei, const float* __restrict__ as_,
                         const float* __restrict__ ad_, float* __restrict__ m,
                         int E_, int ET) {
  int e = blockIdx.x * blockDim.x + threadIdx.x;
  if (e >= ET) return;
  int s = e < E_ ? ei[e] : e - E_;
  int d = e < E_ ? ei[E_ + e] : e - E_;
#pragma unroll
  for (int hd = 0; hd < HEADS; ++hd) {
    float v = leaky(as_[s * HEADS + hd] + ad_[d * HEADS + hd]);
    atomicMaxF(&m[d * HEADS + hd], v);
  }
}

// ---------------------------------- softmax pass 2: exp + segment denominator
__global__ void edge_exp(const int* __restrict__ ei, const float* __restrict__ as_,
                         const float* __restrict__ ad_, const float* __restrict__ m,
                         float* __restrict__ ex, float* __restrict__ denom,
                         int E_, int ET) {
  int e = blockIdx.x * blockDim.x + threadIdx.x;
  if (e >= ET) return;
  int s = e < E_ ? ei[e] : e - E_;
  int d = e < E_ ? ei[E_ + e] : e - E_;
#pragma unroll
  for (int hd = 0; hd < HEADS; ++hd) {
    float v = leaky(as_[s * HEADS + hd] + ad_[d * HEADS + hd]);
    float xv = expf(v - m[d * HEADS + hd]);
    ex[(long long)e * HEADS + hd] = xv;
    atomicAdd(&denom[d * HEADS + hd], xv);
  }
}

// ------------------------------- weighted gather/scatter: one wave per edge
__global__ void edge_aggregate(const float* __restrict__ h, const float* __restrict__ ex,
                               const float* __restrict__ denom,
                               const int* __restrict__ ei,
                               float* __restrict__ out, int E_, int ET) {
  int wid = (int)(((long long)blockIdx.x * blockDim.x + threadIdx.x) >> 5);
  int lane = threadIdx.x & 31;
  if (wid >= ET) return;
  int s = wid < E_ ? ei[wid] : wid - E_;
  int d = wid < E_ ? ei[E_ + wid] : wid - E_;
  int head = lane >> 4;                       // lanes 0-15: ch 0-63 (head 0)
  float alpha = ex[(long long)wid * HEADS + head] / denom[d * HEADS + head];
  float4 v = ((const float4*)(h + (long long)s * HC))[lane];
  float* od = out + (long long)d * HC + lane * 4;
  atomicAdd(od + 0, v.x * alpha);
  atomicAdd(od + 1, v.y * alpha);
  atomicAdd(od + 2, v.z * alpha);
  atomicAdd(od + 3, v.w * alpha);
}

// ------------------------------------------------------------ batchnorm+relu
__global__ void bn_stats(const float* __restrict__ x, float* __restrict__ stats,
                         int n) {
  int c = threadIdx.x;                        // 128 threads
  float s = 0.f, ss = 0.f;
  for (int i = blockIdx.x; i < n; i += gridDim.x) {
    float v = x[(long long)i * HC + c];
    s += v;
    ss += v * v;
  }
  atomicAdd(&stats[c], s);
  atomicAdd(&stats[HC + c], ss);
}

__global__ void bn_finalize(const float* __restrict__ stats,
                            const float* __restrict__ gamma,
                            const float* __restrict__ beta,
                            float* __restrict__ scale, float* __restrict__ shift,
                            int n) {
  int c = threadIdx.x;
  float inv = 1.f / (float)n;
  float mu = stats[c] * inv;
  float var = stats[HC + c] * inv - mu * mu;
  float sc = gamma[c] * rsqrtf(var + EPSBN);
  scale[c] = sc;
  shift[c] = beta[c] - mu * sc;
}

__global__ void bn_apply_relu(float* __restrict__ x, const float* __restrict__ scale,
                              const float* __restrict__ shift, long long total) {
  long long i = (long long)blockIdx.x * blockDim.x + threadIdx.x;
  if (i >= total) return;
  int c = (int)(i & (HC - 1));
  float v = x[i] * scale[c] + shift[c];
  x[i] = v > 0.f ? v : 0.f;
}

// ------------------------------------------------------------------ pooling
__global__ void pool_accum(const float* __restrict__ x, const int* __restrict__ batch,
                           float* __restrict__ pool, float* __restrict__ cnt, int n) {
  long long i = (long long)blockIdx.x * blockDim.x + threadIdx.x;
  if (i >= (long long)n * HC) return;
  int node = (int)(i >> 7);
  int c = (int)(i & (HC - 1));
  int b = batch[node];
  atomicAdd(&pool[b * HC + c], x[i]);
  if (c == 0) atomicAdd(&cnt[b], 1.f);
}

__global__ void final_linear(const float* __restrict__ pool, const float* __restrict__ cnt,
                             const float* __restrict__ Wlin, const float* __restrict__ blin,
                             float* __restrict__ out) {
  int t = threadIdx.x;                        // 128 threads: (b, o)
  if (t >= BGRAPHS * 2) return;
  int b = t >> 1, o = t & 1;
  float inv = 1.f / fmaxf(cnt[b], 1.f);
  float acc = 0.f;
  for (int c = 0; c < HC; ++c)
    acc += pool[b * HC + c] * inv * Wlin[c * 2 + o];
  out[b * 2 + o] = acc + blin[o];
}

// ------------------------------------------------------------------- driver
extern "C" void kernel_launch(void* const* d_in, const int* in_sizes, int n_in,
                              void* d_out, int out_size, void* d_ws, size_t ws_size,
                              hipStream_t stream) {
  const float* x      = (const float*)d_in[0];
  const int*   ei     = (const int*)d_in[1];
  const int*   batch  = (const int*)d_in[2];
  const float* W1     = (const float*)d_in[3];
  const float* asrc1  = (const float*)d_in[4];
  const float* adst1  = (const float*)d_in[5];
  // d_in[6] = b1: cancels exactly inside training-mode BatchNorm -> skipped
  const float* gamma1 = (const float*)d_in[7];
  const float* beta1  = (const float*)d_in[8];
  const float* W2     = (const float*)d_in[9];
  const float* asrc2  = (const float*)d_in[10];
  const float* adst2  = (const float*)d_in[11];
  // d_in[12] = b2: cancels in BN -> skipped
  const float* gamma2 = (const float*)d_in[13];
  const float* beta2  = (const float*)d_in[14];
  const float* Wlin   = (const float*)d_in[15];
  const float* blin   = (const float*)d_in[16];
  float* out = (float*)d_out;

  const int n   = in_sizes[2];          // 50000 nodes
  const int inc = in_sizes[0] / n;      // 512
  const int E_  = in_sizes[1] / 2;      // 800000 edges
  const int ET  = E_ + n;               // + self loops

  // workspace carve-up
  float* ws = (float*)d_ws;
  float* bufA  = ws; ws += (long long)n * HC;       // GEMM output h
  float* bufB  = ws; ws += (long long)n * HC;       // aggregated / activated
  float* as_   = ws; ws += (long long)n * HEADS;
  float* ad_   = ws; ws += (long long)n * HEADS;
  float* mbuf  = ws; ws += (long long)n * HEADS;
  float* denom = ws; ws += (long long)n * HEADS;
  float* exb   = ws; ws += (long long)ET * HEADS;
  float* stats = ws; ws += 2 * HC;
  float* scale = ws; ws += HC;
  float* shift = ws; ws += HC;
  float* pool  = ws; ws += BGRAPHS * HC;
  float* cnt   = ws; ws += BGRAPHS;                 // contiguous with pool

  auto nb = [](long long t, int b) { return (unsigned)((t + b - 1) / b); };

  const float* gemmIn = x;
  int K = inc;
  const float* Ws[2] = {W1, W2};
  const float* As[2] = {asrc1, asrc2};
  const float* Ad[2] = {adst1, adst2};
  const float* Gm[2] = {gamma1, gamma2};
  const float* Bt[2] = {beta1, beta2};

  for (int l = 0; l < 2; ++l) {
    long long mtiles  = (n + 15) / 16;
    long long mblocks = (mtiles + MBLK - 1) / MBLK;
    long long waves   = mblocks * (HC / 16);
    gemm_wmma_f32<<<nb(waves * 32, 256), 256, 0, stream>>>(gemmIn, Ws[l], bufA, n, K, HC);
    node_alpha<<<nb((long long)n * 32, 256), 256, 0, stream>>>(bufA, As[l], Ad[l], as_, ad_, n);
    fill_f32<<<nb((long long)n * HEADS, 256), 256, 0, stream>>>(mbuf, -INFINITY, (long long)n * HEADS);
    fill_f32<<<nb((long long)n * HEADS, 256), 256, 0, stream>>>(denom, 0.f, (long long)n * HEADS);
    edge_max<<<nb(ET, 256), 256, 0, stream>>>(ei, as_, ad_, mbuf, E_, ET);
    edge_exp<<<nb(ET, 256), 256, 0, stream>>>(ei, as_, ad_, mbuf, exb, denom, E_, ET);
    fill_f32<<<nb((long long)n * HC, 256), 256, 0, stream>>>(bufB, 0.f, (long long)n * HC);
    edge_aggregate<<<nb((long long)ET * 32, 256), 256, 0, stream>>>(bufA, exb, denom, ei, bufB, E_, ET);
    fill_f32<<<1, 256, 0, stream>>>(stats, 0.f, 2 * HC);
    bn_stats<<<512, HC, 0, stream>>>(bufB, stats, n);
    bn_finalize<<<1, HC, 0, stream>>>(stats, Gm[l], Bt[l], scale, shift, n);
    bn_apply_relu<<<nb((long long)n * HC, 256), 256, 0, stream>>>(bufB, scale, shift, (long long)n * HC);
    gemmIn = bufB;
    K = HC;
  }

  fill_f32<<<nb(BGRAPHS * HC + BGRAPHS, 256), 256, 0, stream>>>(pool, 0.f, BGRAPHS * HC + BGRAPHS);
  pool_accum<<<nb((long long)n * HC, 256), 256, 0, stream>>>(bufB, batch, pool, cnt, n);
  final_linear<<<1, 128, 0, stream>>>(pool, cnt, Wlin, blin, out);
}